// GATLinkPredictor_79207786873632
// MI455X (gfx1250) — compile-verified
//
#include <hip/hip_runtime.h>
#include <math.h>

typedef float v2f __attribute__((ext_vector_type(2)));
typedef float v8f __attribute__((ext_vector_type(8)));

#define N_NODES 50000
#define N_EDGES 800000
#define E_TOT   (N_EDGES + N_NODES)   // reference appends one self-loop per node
#define HEADS   4
#define HID     64
#define OUT_DIM 128
#define NEG_SLOPE 0.2f

// ---------------------------------------------------------------- utilities

__global__ void fill_f32(float* __restrict__ p, long n, float v) {
    long i = (long)blockIdx.x * blockDim.x + threadIdx.x;
    if (i < n) p[i] = v;
}

__device__ __forceinline__ void atomicMaxF(float* a, float v) {
    // monotone float-max via int views (works for mixed signs, init = -inf)
    if (v >= 0.0f) atomicMax((int*)a, __float_as_int(v));
    else           atomicMin((unsigned int*)a, __float_as_uint(v));
}

// ------------------------------------------------ fp32 WMMA GEMM  C = A @ W
// A: [N,K] row-major, W: [K,M] row-major, C: [N,M].  One wave = one 16x16 tile.
// V_WMMA_F32_16X16X4_F32:  A tile 16x4 (2 VGPRs), B tile 4x16 (2 VGPRs),
// acc 16x16 f32 (8 VGPRs).  Lane layout per CDNA5 ISA 7.12.2.
__global__ void gemm_wmma_f32(const float* __restrict__ A,
                              const float* __restrict__ W,
                              float* __restrict__ C, int K, int M) {
    const int lane = threadIdx.x & 31;
    const int row0 = blockIdx.x * 16;
    const int col0 = blockIdx.y * 16;
    const int m    = lane & 15;      // row-in-tile for A, col-in-tile for B/C
    const int half = lane >> 4;      // 0: K pair {0,1}, 1: K pair {2,3}

    v8f acc = {};
    const float* arow = A + (long)(row0 + m) * K;

    for (int k = 0; k < K; k += 4) {
        const int kk = k + 2 * half;                 // even -> 8B aligned
        v2f a = *(const v2f*)(arow + kk);            // A[m][kk], A[m][kk+1]
        v2f b;
        b.x = W[(long)kk       * M + col0 + m];      // B[kk  ][n]
        b.y = W[(long)(kk + 1) * M + col0 + m];      // B[kk+1][n]
        acc = __builtin_amdgcn_wmma_f32_16x16x4_f32(
                  false, a, false, b, (short)0, acc, false, false);
    }
#pragma unroll
    for (int r = 0; r < 8; ++r) {
        const int row = row0 + r + 8 * half;         // C: VGPR r -> M=r / r+8
        C[(long)row * M + col0 + m] = acc[r];
    }
}

// -------------------------------------- per-(node, head) attention dot terms
// as[n*heads+hd] = h[n,hd,:] . a_src[hd,:] ; ad likewise.  One wave per pair.
__global__ void node_alpha(const float* __restrict__ h,
                           const float* __restrict__ a_src,
                           const float* __restrict__ a_dst,
                           float* __restrict__ as, float* __restrict__ ad,
                           int heads, int F) {
    const int wid  = (int)((blockIdx.x * blockDim.x + threadIdx.x) >> 5);
    const int lane = threadIdx.x & 31;
    if (wid >= N_NODES * heads) return;
    const int n = wid / heads, hd = wid % heads;
    const float* hp  = h + (long)n * heads * F + (long)hd * F;
    const float* sp  = a_src + (long)hd * F;
    const float* dp  = a_dst + (long)hd * F;
    float ss = 0.0f, sd = 0.0f;
    for (int f = lane; f < F; f += 32) {
        const float v = hp[f];
        ss += v * sp[f];
        sd += v * dp[f];
    }
    for (int o = 16; o; o >>= 1) {
        ss += __shfl_xor(ss, o, 32);
        sd += __shfl_xor(sd, o, 32);
    }
    if (lane == 0) { as[wid] = ss; ad[wid] = sd; }
}

// -------------------------------------------------- softmax pass 1: logit+max
__global__ void edge_logit_max(const long long* __restrict__ ei,
                               const float* __restrict__ as,
                               const float* __restrict__ ad,
                               float* __restrict__ el, float* __restrict__ mx,
                               int heads) {
    const long e = (long)blockIdx.x * blockDim.x + threadIdx.x;
    if (e >= E_TOT) return;
    long s, d;
    if (e < N_EDGES) { s = ei[e]; d = ei[N_EDGES + e]; }
    else             { s = d = e - N_EDGES; }          // self-loop
    for (int hd = 0; hd < heads; ++hd) {
        float l = as[s * heads + hd] + ad[d * heads + hd];
        l = (l > 0.0f) ? l : NEG_SLOPE * l;            // LeakyReLU
        el[e * heads + hd] = l;
        atomicMaxF(&mx[d * heads + hd], l);
    }
}

// ------------------------------------------------ softmax pass 2: exp + denom
__global__ void edge_exp_sum(const long long* __restrict__ ei,
                             float* __restrict__ el,
                             const float* __restrict__ mx,
                             float* __restrict__ den, int heads) {
    const long e = (long)blockIdx.x * blockDim.x + threadIdx.x;
    if (e >= E_TOT) return;
    const long d = (e < N_EDGES) ? (long)ei[N_EDGES + e] : e - N_EDGES;
    for (int hd = 0; hd < heads; ++hd) {
        const float w = __expf(el[e * heads + hd] - mx[d * heads + hd]);
        el[e * heads + hd] = w;
        atomicAdd(&den[d * heads + hd], w);
    }
}

// ------------------------------------------- softmax pass 3: weighted scatter
// One block per edge, one thread per (head, feature); Fp = features/head.
__global__ void edge_aggregate(const long long* __restrict__ ei,
                               const float* __restrict__ h,
                               const float* __restrict__ el,
                               const float* __restrict__ den,
                               float* __restrict__ out, int heads, int Fp) {
    const long e = blockIdx.x;
    const int  t = threadIdx.x;
    long s, d;
    if (e < N_EDGES) { s = ei[e]; d = ei[N_EDGES + e]; }
    else             { s = d = e - N_EDGES; }
    const int hd = t / Fp;
    const float alpha = el[e * heads + hd] / (den[d * heads + hd] + 1e-16f);
    const int FT = heads * Fp;
    atomicAdd(&out[d * FT + t], h[s * FT + t] * alpha);
}

// -------------------------------------------------------------- bias + ELU
__global__ void bias_elu(float* __restrict__ x, const float* __restrict__ b,
                         int F, long n) {
    const long i = (long)blockIdx.x * blockDim.x + threadIdx.x;
    if (i >= n) return;
    const float v = x[i] + b[(int)(i % F)];
    x[i] = (v > 0.0f) ? v : (__expf(v) - 1.0f);
}

// -------------------------------------------- bias + row L2 normalize (128d)
// One wave per node; each lane owns 4 contiguous features.
__global__ void bias_normalize(float* __restrict__ out,
                               const float* __restrict__ b) {
    const int wid  = (int)((blockIdx.x * blockDim.x + threadIdx.x) >> 5);
    const int lane = threadIdx.x & 31;
    if (wid >= N_NODES) return;
    float4 v = *(float4*)(out + (long)wid * OUT_DIM + lane * 4);
    const float4 bb = *(const float4*)(b + lane * 4);
    v.x += bb.x; v.y += bb.y; v.z += bb.z; v.w += bb.w;
    float ss = v.x * v.x + v.y * v.y + v.z * v.z + v.w * v.w;
    for (int o = 16; o; o >>= 1) ss += __shfl_xor(ss, o, 32);
    const float inv = 1.0f / fmaxf(sqrtf(ss), 1e-12f);
    v.x *= inv; v.y *= inv; v.z *= inv; v.w *= inv;
    *(float4*)(out + (long)wid * OUT_DIM + lane * 4) = v;
}

// ---------------------------------------------------------------- launcher

extern "C" void kernel_launch(void* const* d_in, const int* in_sizes, int n_in,
                              void* d_out, int out_size, void* d_ws, size_t ws_size,
                              hipStream_t stream) {
    const float*     x      = (const float*)d_in[0];
    const long long* ei     = (const long long*)d_in[1];   // int64 [2, E]
    const float*     W1     = (const float*)d_in[2];
    const float*     a_src1 = (const float*)d_in[3];
    const float*     a_dst1 = (const float*)d_in[4];
    const float*     b1     = (const float*)d_in[5];
    const float*     W2     = (const float*)d_in[6];
    const float*     a_src2 = (const float*)d_in[7];
    const float*     a_dst2 = (const float*)d_in[8];
    const float*     b2     = (const float*)d_in[9];
    float* out = (float*)d_out;

    char* ws = (char*)d_ws;
    size_t off = 0;
    auto alloc = [&](size_t bytes) {
        void* p = ws + off;
        off += (bytes + 255) & ~(size_t)255;
        return p;
    };
    float* h1   = (float*)alloc((size_t)N_NODES * 256 * 4);   // x @ W1
    float* out1 = (float*)alloc((size_t)N_NODES * 256 * 4);   // layer-1 output
    float* h2   = (float*)alloc((size_t)N_NODES * 128 * 4);   // out1 @ W2
    float* as1  = (float*)alloc((size_t)N_NODES * HEADS * 4);
    float* ad1  = (float*)alloc((size_t)N_NODES * HEADS * 4);
    float* m1   = (float*)alloc((size_t)N_NODES * HEADS * 4);
    float* den1 = (float*)alloc((size_t)N_NODES * HEADS * 4);
    float* as2  = (float*)alloc((size_t)N_NODES * 4);
    float* ad2  = (float*)alloc((size_t)N_NODES * 4);
    float* m2   = (float*)alloc((size_t)N_NODES * 4);
    float* den2 = (float*)alloc((size_t)N_NODES * 4);
    float* el1  = (float*)alloc((size_t)E_TOT * HEADS * 4);
    float* el2  = (float*)alloc((size_t)E_TOT * 4);

    const int TB = 256;
    const long n1 = (long)N_NODES * 256;
    const long n2 = (long)N_NODES * 128;
    const int  egrid = (int)((E_TOT + TB - 1) / TB);

    // ---------------- layer 1 ----------------
    fill_f32<<<(int)((n1 + TB - 1) / TB), TB, 0, stream>>>(out1, n1, 0.0f);
    fill_f32<<<(N_NODES * HEADS + TB - 1) / TB, TB, 0, stream>>>(m1, N_NODES * HEADS, -INFINITY);
    fill_f32<<<(N_NODES * HEADS + TB - 1) / TB, TB, 0, stream>>>(den1, N_NODES * HEADS, 0.0f);

    dim3 g1(N_NODES / 16, 256 / 16);
    gemm_wmma_f32<<<g1, 32, 0, stream>>>(x, W1, h1, 128, 256);

    node_alpha<<<(N_NODES * HEADS + 3) / 4, 128, 0, stream>>>(h1, a_src1, a_dst1,
                                                              as1, ad1, HEADS, HID);
    edge_logit_max<<<egrid, TB, 0, stream>>>(ei, as1, ad1, el1, m1, HEADS);
    edge_exp_sum<<<egrid, TB, 0, stream>>>(ei, el1, m1, den1, HEADS);
    edge_aggregate<<<E_TOT, HEADS * HID, 0, stream>>>(ei, h1, el1, den1, out1,
                                                      HEADS, HID);
    bias_elu<<<(int)((n1 + TB - 1) / TB), TB, 0, stream>>>(out1, b1, 256, n1);

    // ---------------- layer 2 ----------------
    fill_f32<<<(int)((n2 + TB - 1) / TB), TB, 0, stream>>>(out, n2, 0.0f);
    fill_f32<<<(N_NODES + TB - 1) / TB, TB, 0, stream>>>(m2, N_NODES, -INFINITY);
    fill_f32<<<(N_NODES + TB - 1) / TB, TB, 0, stream>>>(den2, N_NODES, 0.0f);

    dim3 g2(N_NODES / 16, 128 / 16);
    gemm_wmma_f32<<<g2, 32, 0, stream>>>(out1, W2, h2, 256, 128);

    node_alpha<<<(N_NODES + 3) / 4, 128, 0, stream>>>(h2, a_src2, a_dst2,
                                                      as2, ad2, 1, OUT_DIM);
    edge_logit_max<<<egrid, TB, 0, stream>>>(ei, as2, ad2, el2, m2, 1);
    edge_exp_sum<<<egrid, TB, 0, stream>>>(ei, el2, m2, den2, 1);
    edge_aggregate<<<E_TOT, OUT_DIM, 0, stream>>>(ei, h2, el2, den2, out, 1, OUT_DIM);

    bias_normalize<<<(N_NODES + 7) / 8, 256, 0, stream>>>(out, b2);
}